// LU_19670950216328
// MI455X (gfx1250) — compile-verified
//
#include <hip/hip_runtime.h>

#define DD   128
#define PAD  132           // LDS row stride in floats (bank-conflict avoidance)
#define EPSV 1e-3f
#define NROWS 8192         // B*S = 16*512

typedef __attribute__((ext_vector_type(2))) float v2f;
typedef __attribute__((ext_vector_type(8))) float v8f;

// ---------------------------------------------------------------------------
// Kernel 1: materialize dense U, L and log(diag(U)) from packed triangular
// entries.  tril(k=-1) row-major: idx = i*(i-1)/2 + j.
// triu(k=+1) row-major: idx = 127*i - i*(i-1)/2 + (j-i-1).
// ---------------------------------------------------------------------------
__global__ __launch_bounds__(128) void build_lu(const float* __restrict__ lo,
                                                const float* __restrict__ up,
                                                const float* __restrict__ ud,
                                                float* __restrict__ U,
                                                float* __restrict__ L,
                                                float* __restrict__ logd) {
    int t = blockIdx.x * blockDim.x + threadIdx.x;
    if (t >= DD * DD) return;
    int i = t >> 7, j = t & 127;
    float uv, lv;
    if (i < j) {
        int idx = i * 127 - (i * (i - 1)) / 2 + (j - i - 1);
        uv = up[idx]; lv = 0.0f;
    } else if (i > j) {
        int idx = (i * (i - 1)) / 2 + j;
        uv = 0.0f; lv = lo[idx];
    } else {
        float v = ud[i];
        float sp = (v > 30.0f) ? v : log1pf(expf(v));   // softplus
        sp += EPSV;
        uv = sp; lv = 1.0f;
        logd[i] = logf(sp);
    }
    U[t] = uv;
    L[t] = lv;
}

// ---------------------------------------------------------------------------
// Kernel 2: batched masked LU apply via f32 WMMA.
//   z = U (m*x);  w = L (m*z);  y = m*(w+bias) + (1-m)*x
//   log_det = sum_i m[i] * log(diag_i)
// 128 threads = 4 waves; each wave owns 16 batch rows; 128 blocks cover 8192.
// ---------------------------------------------------------------------------
__global__ __launch_bounds__(128) void lu_apply(const float* __restrict__ x,
                                                const int*   __restrict__ mask,
                                                const float* __restrict__ U,
                                                const float* __restrict__ L,
                                                const float* __restrict__ logd,
                                                const float* __restrict__ bias,
                                                float* __restrict__ y,
                                                float* __restrict__ ldet) {
    __shared__ float sMat[DD * PAD];        // U, then reused for L (66 KB)
    __shared__ float sXm [4 * 16 * PAD];    // per-wave masked-x, then masked-z
    __shared__ float sMk [4 * 16 * PAD];    // per-wave mask as float
    __shared__ float sBias[DD];
    __shared__ float sLogd[DD];

    const int tid  = threadIdx.x;
    const int lane = tid & 31;
    const int wave = tid >> 5;
    const int l15  = lane & 15;
    const int lh   = lane >> 4;                 // lane half: 0 or 1
    const int woff = wave * 16 * PAD;
    const int rowbase = blockIdx.x * 64 + wave * 16;

    // Stage U into LDS (row-major, padded stride).
    for (int idx = tid; idx < DD * DD; idx += 128)
        sMat[(idx >> 7) * PAD + (idx & 127)] = U[idx];
    if (tid < DD) { sBias[tid] = bias[tid]; sLogd[tid] = logd[tid]; }

    // Per-wave tile of masked x and float mask.
    for (int idx = lane; idx < 16 * DD; idx += 32) {
        int r = idx >> 7, c = idx & 127;
        int g = (rowbase + r) * DD + c;
        float mf = (float)mask[g];
        sMk[woff + r * PAD + c] = mf;
        sXm[woff + r * PAD + c] = mf * x[g];
    }
    __syncthreads();

    // ---- Phase 1: Z = Xm * U^T, 8 N-tiles kept live in 64 VGPRs -----------
    v8f acc[8];
#pragma unroll
    for (int t = 0; t < 8; ++t)
#pragma unroll
        for (int e = 0; e < 8; ++e) acc[t][e] = 0.0f;

#pragma unroll
    for (int kc = 0; kc < 32; ++kc) {
        // A fragment: A[l&15][4kc + 2*lh + r]  (shared across the 8 N-tiles)
        v2f a = *(const v2f*)&sXm[woff + l15 * PAD + kc * 4 + 2 * lh];
#pragma unroll
        for (int t = 0; t < 8; ++t) {
            // B fragment: B[k][n] = U[16t + n][k]
            v2f b = *(const v2f*)&sMat[(t * 16 + l15) * PAD + kc * 4 + 2 * lh];
            acc[t] = __builtin_amdgcn_wmma_f32_16x16x4_f32(
                false, a, false, b, (short)0, acc[t], false, false);
        }
    }

    // Mask Z and overwrite the Xm buffer with m*z (per-wave region).
#pragma unroll
    for (int t = 0; t < 8; ++t) {
#pragma unroll
        for (int v = 0; v < 8; ++v) {
            int row = v + 8 * lh;
            int col = t * 16 + l15;
            int idx = woff + row * PAD + col;
            sXm[idx] = acc[t][v] * sMk[idx];
        }
    }
    __syncthreads();

    // Re-stage the matrix buffer with L.
    for (int idx = tid; idx < DD * DD; idx += 128)
        sMat[(idx >> 7) * PAD + (idx & 127)] = L[idx];
    __syncthreads();

    // ---- Phase 2: W = (m*Z) * L^T, fused epilogue per tile ----------------
#pragma unroll
    for (int t = 0; t < 8; ++t) {
        v8f w;
#pragma unroll
        for (int e = 0; e < 8; ++e) w[e] = 0.0f;
#pragma unroll
        for (int kc = 0; kc < 32; ++kc) {
            v2f a = *(const v2f*)&sXm[woff + l15 * PAD + kc * 4 + 2 * lh];
            v2f b = *(const v2f*)&sMat[(t * 16 + l15) * PAD + kc * 4 + 2 * lh];
            w = __builtin_amdgcn_wmma_f32_16x16x4_f32(
                false, a, false, b, (short)0, w, false, false);
        }
#pragma unroll
        for (int v = 0; v < 8; ++v) {
            int row = v + 8 * lh;
            int col = t * 16 + l15;
            float mf = sMk[woff + row * PAD + col];
            int g = (rowbase + row) * DD + col;
            y[g] = mf * (w[v] + sBias[col]) + (1.0f - mf) * x[g];
        }
    }

    // ---- log_det: lanes 0..15 each reduce one row of the mask tile --------
    if (lane < 16) {
        float s = 0.0f;
        const float* mrow = &sMk[woff + lane * PAD];
#pragma unroll 8
        for (int c = 0; c < DD; ++c) s += mrow[c] * sLogd[c];
        ldet[rowbase + lane] = s;
    }
}

// ---------------------------------------------------------------------------
extern "C" void kernel_launch(void* const* d_in, const int* in_sizes, int n_in,
                              void* d_out, int out_size, void* d_ws, size_t ws_size,
                              hipStream_t stream) {
    (void)in_sizes; (void)n_in; (void)out_size; (void)ws_size;

    const float* x    = (const float*)d_in[0];
    const int*   mask = (const int*)  d_in[1];
    const float* lo   = (const float*)d_in[2];
    const float* up   = (const float*)d_in[3];
    const float* ud   = (const float*)d_in[4];
    const float* bias = (const float*)d_in[5];

    float* U    = (float*)d_ws;          // 128*128
    float* L    = U + DD * DD;           // 128*128
    float* logd = L + DD * DD;           // 128

    float* y    = (float*)d_out;                         // 8192*128 f32
    float* ldet = y + (size_t)NROWS * DD;                // 8192 f32
    int*   mout = (int*)(ldet + NROWS);                  // 8192*128 i32 passthrough

    build_lu<<<DD, DD, 0, stream>>>(lo, up, ud, U, L, logd);
    lu_apply<<<NROWS / 64, 128, 0, stream>>>(x, mask, U, L, logd, bias, y, ldet);
    hipMemcpyAsync(mout, mask, (size_t)NROWS * DD * sizeof(int),
                   hipMemcpyDeviceToDevice, stream);
}